// SkipInteractionBlock_1288490189573
// MI455X (gfx1250) — compile-verified
//
#include <hip/hip_runtime.h>
#include <hip/hip_bf16.h>
#include <math.h>

// SkipInteractionBlock for MI455X (gfx1250).
// All dense contractions on V_WMMA_F32_16X16X4_F32 (fp32-exact); messages
// scatter-added with native f32 global atomics into an L2-resident buffer.
// Weights are repacked once per launch into float2 (K,K+1)-interleaved form
// so every WMMA B-fragment is a single global_load_b64.

#define NN 20000
#define NE 320000

typedef __attribute__((ext_vector_type(2))) float v2f;
typedef __attribute__((ext_vector_type(8))) float v8f;

// D(16x16,f32) = A(16x4,f32) x B(4x16,f32) + C
__device__ __forceinline__ v8f wmma_f32_k4(v2f a, v2f b, v8f c) {
  return __builtin_amdgcn_wmma_f32_16x16x4_f32(false, a, false, b,
                                               (short)0, c, false, false);
}

__device__ __forceinline__ float sspf(float x) {
  return fmaxf(x, 0.0f) + log1pf(__expf(-fabsf(x))) - 0.69314718056f;
}

__device__ __forceinline__ void atadd(float* p, float v) {
  __hip_atomic_fetch_add(p, v, __ATOMIC_RELAXED, __HIP_MEMORY_SCOPE_AGENT);
}

// ---------------------------------------------------------------- zero ws
__global__ void k_zero(float4* __restrict__ p, int n4) {
  int i = blockIdx.x * blockDim.x + threadIdx.x;
  const int stride = gridDim.x * blockDim.x;
  for (; i < n4; i += stride) p[i] = make_float4(0.f, 0.f, 0.f, 0.f);
}

// -------------------------------------------------- weight pair-repacking
// dst[kk*64+n] = {src[(2kk)*64+n], src[(2kk+1)*64+n]}  (64-wide matrices)
__global__ void k_pack(const float* __restrict__ src, float2* __restrict__ dst,
                       int npairs) {
  int i = blockIdx.x * blockDim.x + threadIdx.x;
  const int total = npairs * 64;
  const int stride = gridDim.x * blockDim.x;
  for (; i < total; i += stride) {
    const int kk = i >> 6, n = i & 63;
    dst[i] = make_float2(src[(2 * kk) * 64 + n], src[(2 * kk + 1) * 64 + n]);
  }
}

// Shared K-loop: KC chunks of 4, B from packed float2 weights.
template <bool ACONTIG, int KC>
__device__ __forceinline__ void gemm_loop(const float* __restrict__ arow,
                                          const float2* __restrict__ Wp,
                                          int kh, int l16, v8f acc[4]) {
  #pragma unroll 4
  for (int kc = 0; kc < KC; ++kc) {
    const int k0 = kc * 4 + 2 * kh;           // even
    v2f a;
    if (ACONTIG) {
      const float2 av = *(const float2*)(arow + k0);   // global_load_b64
      a.x = av.x; a.y = av.y;
    } else {
      a.x = arow[k0 * 3];                      // vector channel, stride 3
      a.y = arow[(k0 + 1) * 3];
    }
    const float2* bp = Wp + (k0 >> 1) * 64 + l16;
    #pragma unroll
    for (int ct = 0; ct < 4; ++ct) {
      const float2 bv = bp[ct * 16];           // global_load_b64
      v2f b; b.x = bv.x; b.y = bv.y;
      acc[ct] = wmma_f32_k4(a, b, acc[ct]);
    }
  }
}

// ------------------------------------------------- node pre-transform (WMMA)
// s' = s @ W1s / 8 ; v'_d = v_d @ W1v / 8, staged into d_out (input layout).
__global__ void k_pre(const float* __restrict__ nf,
                      const float2* __restrict__ W1sp,   // packed 32x64
                      const float2* __restrict__ W1vp,
                      float* __restrict__ sv) {
  const int lane = threadIdx.x & 31;
  const int wid  = threadIdx.x >> 5;
  const int l16  = lane & 15;
  const int kh   = lane >> 4;
  const int row0 = (blockIdx.x * 8 + wid) * 16;
  int rowA = row0 + l16; if (rowA >= NN) rowA = NN - 1;

  for (int tr = 0; tr < 4; ++tr) {             // 0=s, 1..3 = v_x,v_y,v_z
    const int aoff = (tr == 0) ? 0 : (64 + (tr - 1));
    const float* arow = nf + (size_t)rowA * 256 + aoff;
    v8f acc[4] = {};
    if (tr == 0) gemm_loop<true, 16>(arow, W1sp, kh, l16, acc);
    else         gemm_loop<false, 16>(arow, W1vp, kh, l16, acc);

    #pragma unroll
    for (int ct = 0; ct < 4; ++ct) {
      #pragma unroll
      for (int j = 0; j < 8; ++j) {
        const int r = row0 + j + 8 * kh;       // D layout: VGPR j -> M=j/j+8
        if (r < NN) {
          const int c = ct * 16 + l16;
          const int o = (tr == 0) ? (r * 256 + c)
                                  : (r * 256 + 64 + c * 3 + (tr - 1));
          sv[o] = acc[ct][j] * 0.125f;         // 1/sqrt(64)
        }
      }
    }
  }
}

// --------------------------------------- edge messages + scatter (atomics)
__global__ void k_msg(const float* __restrict__ ea,    // E x 4
                      const float* __restrict__ efeat, // E x 8
                      const int*   __restrict__ eidx,  // 2 x E
                      const float* __restrict__ Wm1,   // 8 x 8
                      const float* __restrict__ Wm2,   // 8 x 256
                      const float* __restrict__ sv,    // staged [n][256]
                      float* __restrict__ msmv) {      // [n][512]
  const int e = blockIdx.x * 8 + (threadIdx.x >> 5);
  if (e >= NE) return;
  const int lane = threadIdx.x & 31;

  const int snd = eidx[e];
  const int rcv = eidx[NE + e];
  const float Y0  = ea[e * 4 + 0];
  const float Y1x = ea[e * 4 + 1];
  const float Y1y = ea[e * 4 + 2];
  const float Y1z = ea[e * 4 + 3];

  float f[8];
  #pragma unroll
  for (int q = 0; q < 8; ++q) f[q] = efeat[e * 8 + q];

  float h[8];
  #pragma unroll
  for (int r = 0; r < 8; ++r) {
    float t = 0.f;
    #pragma unroll
    for (int q = 0; q < 8; ++q) t += f[q] * Wm1[q * 8 + r];
    h[r] = sspf(t * 0.35355339059f);
  }

  const int u0 = lane, u1 = lane + 32;
  float w[4][2];
  #pragma unroll
  for (int k = 0; k < 4; ++k) {
    float t0 = 0.f, t1 = 0.f;
    #pragma unroll
    for (int r = 0; r < 8; ++r) {
      t0 += h[r] * Wm2[r * 256 + k * 64 + u0];
      t1 += h[r] * Wm2[r * 256 + k * 64 + u1];
    }
    w[k][0] = t0 * 0.35355339059f;
    w[k][1] = t1 * 0.35355339059f;
  }

  const float* srow = sv + (size_t)snd * 256;
  float* base = msmv + (size_t)rcv * 512;

  #pragma unroll
  for (int t = 0; t < 2; ++t) {
    const int u = (t == 0) ? u0 : u1;
    const float se = srow[u];
    const float vx = srow[64 + u * 3 + 0];
    const float vy = srow[64 + u * 3 + 1];
    const float vz = srow[64 + u * 3 + 2];
    const float dv = vx * Y1x + vy * Y1y + vz * Y1z;
    atadd(base + u,                      w[0][t] * se * Y0);
    atadd(base + 64 + u,                 w[3][t] * dv * 0.57735026919f);
    atadd(base + 128 + u * 3 + 0,        w[1][t] * se * Y1x);
    atadd(base + 128 + u * 3 + 1,        w[1][t] * se * Y1y);
    atadd(base + 128 + u * 3 + 2,        w[1][t] * se * Y1z);
    atadd(base + 128 + (64 + u) * 3 + 0, w[2][t] * vx * Y0);
    atadd(base + 128 + (64 + u) * 3 + 1, w[2][t] * vy * Y0);
    atadd(base + 128 + (64 + u) * 3 + 2, w[2][t] * vz * Y0);
  }
}

// ------------------------------- node post-transform + skip (WMMA, fused)
__global__ void k_post(const float* __restrict__ msmv,   // [n][512]
                       const float* __restrict__ attrs,  // N x 8
                       const float2* __restrict__ W2sp,  // packed 64x64
                       const float2* __restrict__ W2vp,
                       const float2* __restrict__ WskSp, // packed 256x64
                       const float2* __restrict__ WskVp,
                       float* __restrict__ out) {        // N x 256
  __shared__ float lds[8 * 16 * 64];                     // 32 KB
  const int lane = threadIdx.x & 31;
  const int wid  = threadIdx.x >> 5;
  const int l16  = lane & 15;
  const int kh   = lane >> 4;
  const int row0 = (blockIdx.x * 8 + wid) * 16;
  int rowA = row0 + l16; if (rowA >= NN) rowA = NN - 1;
  float* myLds = lds + wid * (16 * 64);

  // skip-GEMM k = u*8 + a (u-major). Per lane only 4 attr values are needed:
  // even chunks use a = 2kh, odd chunks a = 4+2kh. Fold 1/sqrt(512).
  const float* arow8 = attrs + (size_t)rowA * 8;
  const float attE0 = arow8[2 * kh]     * 0.04419417382f;
  const float attE1 = arow8[2 * kh + 1] * 0.04419417382f;
  const float attO0 = arow8[4 + 2 * kh] * 0.04419417382f;
  const float attO1 = arow8[5 + 2 * kh] * 0.04419417382f;

  for (int tr = 0; tr < 4; ++tr) {                 // 0=s, 1..3=v dims
    const float2* W2p  = (tr == 0) ? W2sp : W2vp;
    const float2* Wskp = (tr == 0) ? WskSp : WskVp;
    const int aoff = (tr == 0) ? 0 : (128 + (tr - 1));
    const float* arow = msmv + (size_t)rowA * 512 + aoff;

    // m2 = msmv @ W2 (K=128)
    v8f acc[4] = {};
    if (tr == 0) gemm_loop<true, 32>(arow, W2p, kh, l16, acc);
    else         gemm_loop<false, 32>(arow, W2p, kh, l16, acc);

    // scale to m2 and stage tile to LDS in A-layout [m][u]
    #pragma unroll
    for (int ct = 0; ct < 4; ++ct) {
      #pragma unroll
      for (int j = 0; j < 8; ++j) {
        acc[ct][j] *= 0.08838834765f;              // 1/sqrt(128)
        myLds[(j + 8 * kh) * 64 + ct * 16 + l16] = acc[ct][j];
      }
    }
    __syncthreads();

    // skip GEMM K=512, u-major: one ds_load of m2[m][u] feeds 8 WMMAs.
    #pragma unroll 2
    for (int i = 0; i < 64; ++i) {                 // u = i
      const float m2 = myLds[l16 * 64 + i];
      const float2* bp = Wskp + (4 * i + kh) * 64 + l16;  // kk = k0>>1
      v2f a;
      a.x = m2 * attE0; a.y = m2 * attE1;          // k0 = 8i + 2kh
      #pragma unroll
      for (int ct = 0; ct < 4; ++ct) {
        const float2 bv = bp[ct * 16];
        v2f b; b.x = bv.x; b.y = bv.y;
        acc[ct] = wmma_f32_k4(a, b, acc[ct]);
      }
      a.x = m2 * attO0; a.y = m2 * attO1;          // k0 = 8i + 4 + 2kh
      #pragma unroll
      for (int ct = 0; ct < 4; ++ct) {
        const float2 bv = bp[2 * 64 + ct * 16];
        v2f b; b.x = bv.x; b.y = bv.y;
        acc[ct] = wmma_f32_k4(a, b, acc[ct]);      // acc = m2 + sk
      }
    }

    #pragma unroll
    for (int ct = 0; ct < 4; ++ct) {
      #pragma unroll
      for (int j = 0; j < 8; ++j) {
        const int r = row0 + j + 8 * kh;
        if (r < NN) {
          const int c = ct * 16 + l16;
          const int o = (tr == 0) ? (r * 256 + c)
                                  : (r * 256 + 64 + c * 3 + (tr - 1));
          out[o] = acc[ct][j];
        }
      }
    }
    __syncthreads();
  }
}

extern "C" void kernel_launch(void* const* d_in, const int* in_sizes, int n_in,
                              void* d_out, int out_size, void* d_ws, size_t ws_size,
                              hipStream_t stream) {
  const float* node_attrs = (const float*)d_in[0];
  const float* node_feats = (const float*)d_in[1];
  const float* edge_attrs = (const float*)d_in[2];
  const float* edge_feats = (const float*)d_in[3];
  const int*   edge_index = (const int*)d_in[4];
  const float* W1s  = (const float*)d_in[5];
  const float* W1v  = (const float*)d_in[6];
  const float* Wm1  = (const float*)d_in[7];
  const float* Wm2  = (const float*)d_in[8];
  const float* W2s  = (const float*)d_in[9];
  const float* W2v  = (const float*)d_in[10];
  const float* WskS = (const float*)d_in[11];
  const float* WskV = (const float*)d_in[12];

  float* out  = (float*)d_out;            // N x 256 (staging for s'/v')
  float* msmv = (float*)d_ws;             // N x 512 accumulator (40.96 MB)

  // Packed-weight region after the accumulator (float2 units, 360 KB total).
  float2* packBase = (float2*)(msmv + (size_t)NN * 512);
  float2* W1sp  = packBase;               // 32*64
  float2* W1vp  = packBase + 2048;        // 32*64
  float2* W2sp  = packBase + 4096;        // 64*64
  float2* W2vp  = packBase + 8192;        // 64*64
  float2* WskSp = packBase + 12288;       // 256*64
  float2* WskVp = packBase + 28672;       // 256*64

  k_pack<<<8,  256, 0, stream>>>(W1s,  W1sp,  32);
  k_pack<<<8,  256, 0, stream>>>(W1v,  W1vp,  32);
  k_pack<<<16, 256, 0, stream>>>(W2s,  W2sp,  64);
  k_pack<<<16, 256, 0, stream>>>(W2v,  W2vp,  64);
  k_pack<<<64, 256, 0, stream>>>(WskS, WskSp, 256);
  k_pack<<<64, 256, 0, stream>>>(WskV, WskVp, 256);
  k_zero<<<2048, 256, 0, stream>>>((float4*)msmv, NN * 512 / 4);
  k_pre <<<(NN + 127) / 128, 256, 0, stream>>>(node_feats, W1sp, W1vp, out);
  k_msg <<<(NE + 7) / 8, 256, 0, stream>>>(edge_attrs, edge_feats, edge_index,
                                           Wm1, Wm2, out, msmv);
  k_post<<<(NN + 127) / 128, 256, 0, stream>>>(msmv, node_attrs, W2sp, W2vp,
                                               WskSp, WskVp, out);
}